// Net_OnOff_85890755985947
// MI455X (gfx1250) — compile-verified
//
#include <hip/hip_runtime.h>

typedef __attribute__((ext_vector_type(2))) float v2f;
typedef __attribute__((ext_vector_type(8))) float v8f;

#define CHUNK 256

#define BETA_LIF_F  0.98f
#define BETA_GAIN_F 0.9f
#define BETA_EPSC_F 0.9f
#define BETA_TDE_F  0.98f
#define THR_F       1.0f

__device__ __forceinline__ int imin(int a, int b) { return a < b ? a : b; }

// Single workgroup, 2 wave32 waves:
//   wave 0 : serial LIF/TDE recurrence (latency-bound chain), LDS-only I/O
//   wave 1 : async-to-LDS input prefetch + WMMA precompute of A = x @ We^T
//            + coalesced output drain (double-buffered, 1 barrier/chunk)
__global__ __launch_bounds__(64)
void snn_tde_scan(const float* __restrict__ x,   // T*2
                  const float* __restrict__ We,  // 2x2 row-major
                  const float* __restrict__ Wi,  // 2x2 row-major
                  float* __restrict__ out,       // T*10
                  int T)
{
    __shared__ __align__(16) float s_a[2][CHUNK * 2];        // staged A (double buffer)
    __shared__ __align__(16) float s_x[CHUNK * 2];           // async-load target
    __shared__ __align__(16) float s_out[2][CHUNK * 10 + 32];// staged outputs + dump

    const int tid  = threadIdx.x;
    const int lane = tid & 31;
    const int wave = tid >> 5;
    const int NC   = (T + CHUNK - 1) / CHUNK;

    // ---- wave-0 scan state (registers, carried across all chunks) ----
    float m0 = 0.f, m1 = 0.f, sp0 = 0.f, sp1 = 0.f;
    float g = 0.f, e = 0.f, tmv = 0.f, tsv = 0.f;
    const float wi00 = Wi[0], wi01 = Wi[1], wi10 = Wi[2], wi11 = Wi[3];
    const bool l1 = (lane == 1), l2 = (lane == 2), l3 = (lane == 3),
               l4 = (lane == 4), l5 = (lane == 5), l6 = (lane == 6),
               l7 = (lane == 7), l8 = (lane == 8), l9 = (lane == 9);
    const bool lt10 = (lane < 10);

    // ---- wave-1 WMMA B operand: B[k][n] = (We^T)[k][n] = We[n][k], K>=2 pad 0
    const int  n16 = lane & 15;
    const bool lo  = (lane < 16);
    const float we00 = We[0], we01 = We[1], we10 = We[2], we11 = We[3];
    v2f bmat;
    bmat.x = lo ? (n16 == 0 ? we00 : (n16 == 1 ? we10 : 0.f)) : 0.f; // K=0 row
    bmat.y = lo ? (n16 == 0 ? we01 : (n16 == 1 ? we11 : 0.f)) : 0.f; // K=1 row

    for (int c = -1; c < NC; ++c) {
        __syncthreads();
        if (wave == 0) {
            if (c >= 0) {
                const float* ap = &s_a[c & 1][0];
                float* ob = &s_out[c & 1][0];
                const int steps = imin(CHUNK, T - c * CHUNK);
                #pragma unroll 2
                for (int i = 0; i < steps; ++i) {
                    const float a0 = ap[2 * i + 0];
                    const float a1 = ap[2 * i + 1];
                    const float cur0 = a0 + sp0 * wi00 + sp1 * wi01;
                    const float cur1 = a1 + sp0 * wi10 + sp1 * wi11;
                    const float m0n = BETA_LIF_F * m0 + cur0 - sp0;
                    const float m1n = BETA_LIF_F * m1 + cur1 - sp1;
                    const float s0n = (m0n - THR_F > 0.f) ? 1.f : 0.f;
                    const float s1n = (m1n - THR_F > 0.f) ? 1.f : 0.f;
                    // g*0.9*(1-fac)+fac with fac in {0,1}
                    const float gn  = (s0n > 0.5f) ? 1.f : BETA_GAIN_F * g;
                    const float en  = BETA_EPSC_F * e + gn * s1n;      // trg in {0,1}
                    const float tmn = BETA_TDE_F * tmv + en - tsv;
                    const float tsn = (tmn - THR_F > 0.f) ? 1.f : 0.f;
                    // lane c<10 stages component c; lanes>=10 hit dump slots
                    float v = cur0;
                    v = l1 ? cur1 : v; v = l2 ? m0n : v; v = l3 ? m1n : v;
                    v = l4 ? s0n  : v; v = l5 ? s1n : v; v = l6 ? tsn : v;
                    v = l7 ? gn   : v; v = l8 ? en  : v; v = l9 ? tmn : v;
                    ob[(lt10 ? i * 10 : CHUNK * 10) + lane] = v;
                    m0 = m0n; m1 = m1n; sp0 = s0n; sp1 = s1n;
                    g = gn; e = en; tmv = tmn; tsv = tsn;
                }
            }
        } else {
            const int nc = c + 1;
            // (1) async prefetch x chunk nc into LDS (gfx1250 ASYNCcnt path)
            if (nc < NC) {
                const char* gsrc = (const char*)(x + (size_t)nc * (CHUNK * 2));
                const int nbytes = imin(CHUNK, T - nc * CHUNK) * 8;
                const unsigned lbase = (unsigned)(size_t)(&s_x[0]);
                #pragma unroll
                for (int j = 0; j < (CHUNK * 8) / (32 * 16); ++j) {
                    const int off = (j * 32 + lane) * 16;
                    if (off < nbytes) {
                        unsigned la = lbase + (unsigned)off;
                        const char* ga = gsrc + off;
                        asm volatile("global_load_async_to_lds_b128 %0, %1, off"
                                     :: "v"(la), "v"(ga)
                                     : "memory");
                    }
                }
            }
            // (2) drain previous chunk's staged outputs while the async load flies
            if (c >= 1) {
                const int cc = c - 1;
                const int stepsc = imin(CHUNK, T - cc * CHUNK);
                const float* src = &s_out[cc & 1][0];
                float* dst = out + (size_t)cc * (CHUNK * 10);
                const int npairs = stepsc * 5;
                for (int p = lane; p < npairs; p += 32) {
                    *(v2f*)(dst + 2 * p) = *(const v2f*)(src + 2 * p);
                }
            }
            // (3) wait async, then matrix-precompute A = x @ We^T on the matrix unit
            if (nc < NC) {
                asm volatile("s_wait_asynccnt 0x0" ::: "memory");
                float* adst = &s_a[nc & 1][0];
                for (int tt = 0; tt < CHUNK / 16; ++tt) {
                    const int tl = tt * 16 + n16;
                    const v2f xv = *(const v2f*)&s_x[2 * tl];
                    v2f amat;                       // A 16x4: lanes<16 K=0..1, pad K=2..3 = 0
                    amat.x = lo ? xv.x : 0.f;
                    amat.y = lo ? xv.y : 0.f;
                    v8f cm = {};
#if __has_builtin(__builtin_amdgcn_wmma_f32_16x16x4_f32)
                    cm = __builtin_amdgcn_wmma_f32_16x16x4_f32(
                        false, amat, false, bmat, (short)0, cm, false, false);
#else
                    // exact VALU fallback (only lanes n16<2 results are consumed)
                    {
                        const float wn0 = (n16 == 0) ? we00 : we10;
                        const float wn1 = (n16 == 0) ? we01 : we11;
                        #pragma unroll
                        for (int r = 0; r < 8; ++r) {
                            const int row = tt * 16 + r + (lo ? 0 : 8);
                            cm[r] = s_x[2 * row] * wn0 + s_x[2 * row + 1] * wn1;
                        }
                    }
#endif
                    // C 16x16 f32 layout: VGPR r -> M = r (+8 for hi half), N = lane%16
                    if (n16 < 2) {
                        const int mbase = tt * 16 + (lo ? 0 : 8);
                        #pragma unroll
                        for (int r = 0; r < 8; ++r) {
                            adst[(mbase + r) * 2 + n16] = cm[r];
                        }
                    }
                }
            }
        }
    }

    // epilogue: drain the last chunk's staged outputs with all 64 threads
    __syncthreads();
    if (NC > 0) {
        const int cc = NC - 1;
        const int stepsc = T - cc * CHUNK;
        const float* src = &s_out[cc & 1][0];
        float* dst = out + (size_t)cc * (CHUNK * 10);
        const int npairs = stepsc * 5;
        for (int p = tid; p < npairs; p += 64) {
            *(v2f*)(dst + 2 * p) = *(const v2f*)(src + 2 * p);
        }
    }
}

extern "C" void kernel_launch(void* const* d_in, const int* in_sizes, int n_in,
                              void* d_out, int out_size, void* d_ws, size_t ws_size,
                              hipStream_t stream) {
    const float* x  = (const float*)d_in[0];  // (T,1,2) f32
    const float* We = (const float*)d_in[1];  // (2,2)   f32
    const float* Wi = (const float*)d_in[2];  // (2,2)   f32
    float* out = (float*)d_out;               // (T,1,10) f32
    const int T = in_sizes[0] / 2;
    snn_tde_scan<<<dim3(1), dim3(64), 0, stream>>>(x, We, Wi, out, T);
}